// DeformConv2d_62285615727271
// MI455X (gfx1250) — compile-verified
//
#include <hip/hip_runtime.h>
#include <hip/hip_bf16.h>

typedef float v2f __attribute__((ext_vector_type(2)));
typedef float v8f __attribute__((ext_vector_type(8)));

#define C_IN   64
#define O_OUT  64
#define KTAPS  9
#define HW     128

// LDS layout (float offsets) -- 52,224 B total => ~4 workgroups resident / WGP
//   Wt : current tap weights [c2(32)][o(64)][2] = 4096 floats (16384 B)
//   S2 : sampled matrix      [c2(32)][p(128)][2] = 8192 floats (32768 B)
//   coords: iy0[128](int) ix0[128](int) w00/w01/w10/w11[128](float) = 3072 B
#define WT_OFF 0
#define S2_OFF 4096
#define CO_OFF (4096 + 8192)

extern "C" __global__ __launch_bounds__(256)
void deform_conv2d_wmma_f32(const float* __restrict__ x,
                            const float* __restrict__ offset,
                            const float* __restrict__ weight,
                            float* __restrict__ out)
{
    extern __shared__ float lds[];
    float* Wt   = lds + WT_OFF;
    float* S2   = lds + S2_OFF;
    int*   iy0a = (int*)(lds + CO_OFF);
    int*   ix0a = iy0a + 128;
    float* w00a = (float*)(ix0a + 128);
    float* w01a = w00a + 128;
    float* w10a = w01a + 128;
    float* w11a = w10a + 128;

    const int t    = threadIdx.x;
    const int lane = t & 31;
    const int wv   = t >> 5;
    const int n    = blockIdx.x >> 7;   // image
    const int row  = blockIdx.x & 127;  // output row

    // per-wave: 16 output channels x 64 pixels = 4 accumulator tiles
    v8f acc0 = {}, acc1 = {}, acc2 = {}, acc3 = {};

    const int m     = lane & 15;          // M (A) == N (B) lane index
    const int hi    = lane >> 4;          // selects K pair 0/1 vs 2/3
    const int obase = (wv & 3) * 16;      // output-channel tile
    const int pgrp  = (wv >> 2) * 4;      // pixel-tile group: tiles 0-3 or 4-7

    const int p   = t & 127;              // pixel owned in gather phase
    const int ch0 = t >> 7;               // channel parity in gather phase

    // weight staging indices (per tap): thread loads 16 elements, c-fastest
    const int wc0 = t & 63;               // input channel
    const int wo0 = t >> 6;               // output channel base (step 4)

    for (int k = 0; k < KTAPS; ++k) {
        __syncthreads();   // previous tap finished reading S2 / Wt / coords

        // ---- stage tap-k weights into LDS:  (O,C,3,3) -> Wt[c/2][o][2] ----
        #pragma unroll
        for (int j = 0; j < 16; ++j) {
            int o = wo0 + 4 * j;
            float w = weight[((size_t)o * C_IN + wc0) * KTAPS + k];
            Wt[(((wc0 >> 1) * 64 + o) << 1) + (wc0 & 1)] = w;
        }

        // ---- per-pixel bilinear coords for this tap (shared by all c) ----
        if (t < 128) {
            int ky = k / 3, kx = k - 3 * ky;
            float oy = offset[(((size_t)n * 18 + 2 * k    ) * 128 + row) * 128 + t];
            float ox = offset[(((size_t)n * 18 + 2 * k + 1) * 128 + row) * 128 + t];
            float py = (float)(row - 1 + ky) + oy;
            float px = (float)(t   - 1 + kx) + ox;
            float y0f = floorf(py), x0f = floorf(px);
            float wy = py - y0f,    wx = px - x0f;
            iy0a[t] = (int)y0f;
            ix0a[t] = (int)x0f;
            w00a[t] = (1.f - wy) * (1.f - wx);
            w01a[t] = (1.f - wy) * wx;
            w10a[t] = wy * (1.f - wx);
            w11a[t] = wy * wx;
        }
        __syncthreads();

        // ---- gather stage: fill S2[c/2][p][2] ----
        {
            int y0 = iy0a[p], x0 = ix0a[p];
            int y1 = y0 + 1,  x1 = x0 + 1;
            float w00 = w00a[p], w01 = w01a[p], w10 = w10a[p], w11 = w11a[p];
            bool ry0 = (y0 >= 0) & (y0 < HW);
            bool ry1 = (y1 >= 0) & (y1 < HW);
            bool rx0 = (x0 >= 0) & (x0 < HW);
            bool rx1 = (x1 >= 0) & (x1 < HW);
            int yc0 = min(max(y0, 0), HW - 1), yc1 = min(max(y1, 0), HW - 1);
            int xc0 = min(max(x0, 0), HW - 1), xc1 = min(max(x1, 0), HW - 1);
            int i00 = yc0 * HW + xc0, i01 = yc0 * HW + xc1;
            int i10 = yc1 * HW + xc0, i11 = yc1 * HW + xc1;
            float f00 = (ry0 & rx0) ? w00 : 0.f;
            float f01 = (ry0 & rx1) ? w01 : 0.f;
            float f10 = (ry1 & rx0) ? w10 : 0.f;
            float f11 = (ry1 & rx1) ? w11 : 0.f;
            const float* xb = x + (size_t)n * C_IN * (HW * HW);
            #pragma unroll 4
            for (int i = 0; i < 32; ++i) {
                int c = 2 * i + ch0;
                const float* xc = xb + (size_t)c * (HW * HW);
                float v = xc[i00] * f00 + xc[i01] * f01
                        + xc[i10] * f10 + xc[i11] * f11;
                S2[(((c >> 1) * 128 + p) << 1) + (c & 1)] = v;
            }
        }
        __syncthreads();

        // ---- WMMA: accumulate W_k[16x64] * S_k[64x64] per wave ----
        #pragma unroll
        for (int s = 0; s < 16; ++s) {
            int c2 = 2 * s + hi;   // K pair index: lanes 0-15 -> K 0,1; 16-31 -> K 2,3
            v2f a = *(const v2f*)(Wt + ((c2 * 64 + m) << 1));
            const float* Sr = S2 + ((c2 * 128) << 1);
            v2f b0 = *(const v2f*)(Sr + (((pgrp + 0) * 16 + m) << 1));
            v2f b1 = *(const v2f*)(Sr + (((pgrp + 1) * 16 + m) << 1));
            v2f b2 = *(const v2f*)(Sr + (((pgrp + 2) * 16 + m) << 1));
            v2f b3 = *(const v2f*)(Sr + (((pgrp + 3) * 16 + m) << 1));
            acc0 = __builtin_amdgcn_wmma_f32_16x16x4_f32(false, a, false, b0, (short)0, acc0, false, false);
            acc1 = __builtin_amdgcn_wmma_f32_16x16x4_f32(false, a, false, b1, (short)0, acc1, false, false);
            acc2 = __builtin_amdgcn_wmma_f32_16x16x4_f32(false, a, false, b2, (short)0, acc2, false, false);
            acc3 = __builtin_amdgcn_wmma_f32_16x16x4_f32(false, a, false, b3, (short)0, acc3, false, false);
        }
    }

    // ---- store accumulators (C/D layout: VGPR r -> M = r + 8*hi, N = lane&15) ----
    const int col16 = lane & 15;
    float* ob = out + (((size_t)n * O_OUT + obase) * 128 + row) * 128;
    #pragma unroll
    for (int r = 0; r < 8; ++r) {
        size_t orow = (size_t)(r + 8 * hi) * (size_t)(HW * HW);
        ob[orow + (pgrp + 0) * 16 + col16] = acc0[r];
        ob[orow + (pgrp + 1) * 16 + col16] = acc1[r];
        ob[orow + (pgrp + 2) * 16 + col16] = acc2[r];
        ob[orow + (pgrp + 3) * 16 + col16] = acc3[r];
    }
}

extern "C" void kernel_launch(void* const* d_in, const int* in_sizes, int n_in,
                              void* d_out, int out_size, void* d_ws, size_t ws_size,
                              hipStream_t stream) {
    (void)in_sizes; (void)n_in; (void)out_size; (void)d_ws; (void)ws_size;
    const float* x      = (const float*)d_in[0];
    const float* offset = (const float*)d_in[1];
    const float* weight = (const float*)d_in[2];
    float* out = (float*)d_out;

    // LDS: tap weights 16384 B + S2 32768 B + coords 3072 B = 52224 B
    const size_t shmem = (size_t)(4096 + 8192) * 4 + 128 * 4 * 2 + 128 * 4 * 4;
    dim3 grid(8 * 128);   // one block per (image, output row)
    dim3 block(256);      // 8 wave32
    hipLaunchKernelGGL(deform_conv2d_wmma_f32, grid, block, shmem, stream,
                       x, offset, weight, out);
}